// CrossHeadProjection_9019431322322
// MI455X (gfx1250) — compile-verified
//
#include <hip/hip_runtime.h>

typedef float v2f __attribute__((ext_vector_type(2)));
typedef float v4f __attribute__((ext_vector_type(4)));
typedef float v8f __attribute__((ext_vector_type(8)));

#define B_ 2
#define H_ 16
#define T_ 1024
#define S_ 1024
#define G_ 2
#define M_ 8
#define I_ 2

#define TT 16                      // t iterations per wave
#define SBLK (S_ / 16)             // 64 s-blocks
#define TCH (T_ / TT)              // 64 t-chunks
#define TOTAL_WAVES (B_ * SBLK * TCH)   // 8192 waves

__global__ __launch_bounds__(256) void chp_wmma_kernel(
    const float* __restrict__ x,
    const float* __restrict__ qw1, const float* __restrict__ qw2,
    const float* __restrict__ kw1, const float* __restrict__ kw2,
    const float* __restrict__ qdd, const float* __restrict__ kdd,
    float* __restrict__ out)
{
    const int wid = (blockIdx.x * blockDim.x + threadIdx.x) >> 5;
    if (wid >= TOTAL_WAVES) return;          // whole-wave uniform: EXEC stays all-1s
    const int lane = threadIdx.x & 31;
    const int half = lane >> 4;              // 0: M=0..7 block, 1: M=8..15 block (C layout)
    const int col  = lane & 15;              // N (= s within block) / M (A layout)

    const int b    = wid / (SBLK * TCH);
    const int rem  = wid % (SBLK * TCH);
    const int sblk = rem % SBLK;
    const int tch  = rem / SBLK;
    const int s    = sblk * 16 + col;
    const int t0   = tch * TT;

    // ---- source-side (per-s) weights: loaded once, reused over the t-loop ----
    const int g = half;                      // group this lane's C rows belong to
    const float* k1p = kw1 + (((size_t)b * S_ + s) * G_ + g) * (M_ * I_);
    const float* k2p = kw2 + (((size_t)b * S_ + s) * G_ + g) * (M_ * I_);
    const float* kdp = kdd + (((size_t)b * S_ + s) * G_ + g) * M_;
    float ka0[M_], ka1[M_], kb0[M_], kb1[M_], kd[M_];
#pragma unroll
    for (int m = 0; m < M_; m += 2) {        // 16B-aligned b128 loads
        v4f a = *(const v4f*)(k1p + m * 2);
        v4f c = *(const v4f*)(k2p + m * 2);
        ka0[m] = a.x; ka1[m] = a.y; ka0[m + 1] = a.z; ka1[m + 1] = a.w;
        kb0[m] = c.x; kb1[m] = c.y; kb0[m + 1] = c.z; kb1[m + 1] = c.w;
    }
    {
        v4f d0 = *(const v4f*)(kdp);
        v4f d1 = *(const v4f*)(kdp + 4);
        kd[0] = d0.x; kd[1] = d0.y; kd[2] = d0.z; kd[3] = d0.w;
        kd[4] = d1.x; kd[5] = d1.y; kd[6] = d1.z; kd[7] = d1.w;
    }

    const int gM = col >> 3;                 // A-layout: this lane supplies row M=col of Wq
    const int mM = col & 7;
    // branch-free block-diagonal mask: gK is an unroll-time constant, so this is one v_mul
    const float selg[2] = { (gM == 0) ? 1.f : 0.f, (gM == 1) ? 1.f : 0.f };
    const size_t plane = (size_t)T_ * S_;    // stride between h-planes of x

    for (int tt = 0; tt < TT; ++tt) {
        const int t = t0 + tt;
        const float* q1t = qw1 + (((size_t)b * T_ + t) * G_) * (M_ * I_);
        const float* q2t = qw2 + (((size_t)b * T_ + t) * G_) * (M_ * I_);
        const v2f q2 = *(const v2f*)(q2t + gM * (M_ * I_) + mM * I_);

        v8f C = {0.f, 0.f, 0.f, 0.f, 0.f, 0.f, 0.f, 0.f};
        const float* xb = x + (((size_t)b * H_) * T_ + t) * S_ + s;

        // ret_q[16,16] = Wq(t)[16x16] @ x[16,16]  via 4 chained f32 WMMAs (K=4 each)
#pragma unroll
        for (int kbI = 0; kbI < 4; ++kbI) {
            const int gK  = kbI >> 1;                 // group of this K-slice (block-diag Wq)
            const int k0  = 4 * kbI + 2 * half;       // A layout: v0 holds K=2*half of slice
            const int mK0 = k0 & 7;                   // even -> one aligned b128 covers both pairs
            v4f q1 = *(const v4f*)(q1t + gK * (M_ * I_) + mK0 * I_);
            const float sel = selg[gK];
            v2f A;
            A.x = sel * (q2.x * q1.x + q2.y * q1.y);
            A.y = sel * (q2.x * q1.z + q2.y * q1.w);
            // B layout (4x16): v0 -> K = half, v1 -> K = 2+half of slice; N = col
            v2f Bv;
            Bv.x = xb[(size_t)(4 * kbI + half) * plane];
            Bv.y = xb[(size_t)(4 * kbI + 2 + half) * plane];
            C = __builtin_amdgcn_wmma_f32_16x16x4_f32(
                    false, A, false, Bv, (short)0, C, false, false);
        }

        // ---- epilogue in C/D layout: lane owns (h = r + 8*half, s), r = 0..7 ----
        const float* xc = x + (((size_t)b * H_ + 8 * half) * T_ + t) * S_ + s;
        float xv[8];
#pragma unroll
        for (int r = 0; r < 8; ++r) xv[r] = xc[(size_t)r * plane];

        if (tt + 1 < TT) __builtin_prefetch(xc + S_, 0, 1);   // next t row -> global_prefetch

        const float* qdp = qdd + (((size_t)b * T_ + t) * G_ + half) * M_;
        float qd[8];
        {
            v4f d0 = *(const v4f*)(qdp);
            v4f d1 = *(const v4f*)(qdp + 4);
            qd[0] = d0.x; qd[1] = d0.y; qd[2] = d0.z; qd[3] = d0.w;
            qd[4] = d1.x; qd[5] = d1.y; qd[6] = d1.z; qd[7] = d1.w;
        }

        // source-side rank-2 hidden (contraction over this lane's 8 in-group heads)
        float hk0 = 0.f, hk1 = 0.f;
#pragma unroll
        for (int r = 0; r < 8; ++r) { hk0 += ka0[r] * xv[r]; hk1 += ka1[r] * xv[r]; }

        float* op = out + (((size_t)b * H_ + 8 * half) * T_ + t) * S_ + s;
#pragma unroll
        for (int r = 0; r < 8; ++r) {
            float res = C[r]
                      + xv[r] * (1.f + qd[r] + kd[r])
                      + kb0[r] * hk0 + kb1[r] * hk1;
            op[(size_t)r * plane] = res;
        }
    }
}

extern "C" void kernel_launch(void* const* d_in, const int* in_sizes, int n_in,
                              void* d_out, int out_size, void* d_ws, size_t ws_size,
                              hipStream_t stream) {
    (void)in_sizes; (void)n_in; (void)d_ws; (void)ws_size; (void)out_size;
    const float* x   = (const float*)d_in[0];
    const float* qw1 = (const float*)d_in[1];
    const float* qw2 = (const float*)d_in[2];
    const float* kw1 = (const float*)d_in[3];
    const float* kw2 = (const float*)d_in[4];
    const float* qdd = (const float*)d_in[5];
    const float* kdd = (const float*)d_in[6];
    float* out = (float*)d_out;

    const int threads = 256;                          // 8 waves per block
    const int blocks  = (TOTAL_WAVES * 32) / threads; // 1024 blocks
    chp_wmma_kernel<<<blocks, threads, 0, stream>>>(x, qw1, qw2, kw1, kw2, qdd, kdd, out);
}